// SimpleGNN_60327110640120
// MI455X (gfx1250) — compile-verified
//
#include <hip/hip_runtime.h>

typedef float v2f __attribute__((ext_vector_type(2)));
typedef float v8f __attribute__((ext_vector_type(8)));

#define NNODES 100000
#define NEDGES 3200000LL
#define HID 128
#define SXS 132   // LDS row stride (floats): bank = (4*m + k) % 64 -> conflict-free

// ---------------- utility ----------------
__global__ __launch_bounds__(256) void zero_kernel(float* __restrict__ p, long long n) {
    long long i = (long long)blockIdx.x * 256 + threadIdx.x;
    if (i < n) p[i] = 0.0f;
}

__global__ __launch_bounds__(256) void deg_kernel(const long long* __restrict__ dst,
                                                  const float* __restrict__ ew,
                                                  float* __restrict__ deg, long long nE) {
    long long e = (long long)blockIdx.x * 256 + threadIdx.x;
    if (e < nE) atomicAdd(&deg[(int)dst[e]], ew[e]);
}

__global__ __launch_bounds__(256) void dinv_kernel(const float* __restrict__ deg,
                                                   float* __restrict__ dinv, int n) {
    int i = blockIdx.x * 256 + threadIdx.x;
    if (i < n) {
        float d = deg[i];
        dinv[i] = (d > 0.0f) ? rsqrtf(d) : 0.0f;
    }
}

__global__ __launch_bounds__(256) void norm_kernel(const long long* __restrict__ src,
                                                   const long long* __restrict__ dst,
                                                   const float* __restrict__ ew,
                                                   const float* __restrict__ dinv,
                                                   float* __restrict__ nrm, long long nE) {
    long long e = (long long)blockIdx.x * 256 + threadIdx.x;
    if (e < nE) nrm[e] = dinv[(int)src[e]] * ew[e] * dinv[(int)dst[e]];
}

__global__ __launch_bounds__(256) void init_bias_kernel(float* __restrict__ out,
                                                        const float* __restrict__ b, long long n) {
    long long i = (long long)blockIdx.x * 256 + threadIdx.x;
    if (i < n) out[i] = b[(int)(i & (HID - 1))];
}

// ---------------- WMMA fp32 GEMM: H = X @ W  (X: nrows x 128, W: 128 x 128) ----------------
// Block: 256 threads = 8 waves; block computes rows [row0, row0+16), wave w computes cols [16w,16w+16).
__global__ __launch_bounds__(256) void gemm128_wmma(const float* __restrict__ X,
                                                    const float* __restrict__ Wm,
                                                    float* __restrict__ H, int nrows) {
    __shared__ float sX[16 * SXS];        // X tile, padded
    __shared__ float sWt[HID * SXS];      // W transposed: sWt[n*SXS + k] = W[k][n]

    const int t = threadIdx.x;
    const int row0 = blockIdx.x * 16;
    const bool full = (row0 + 16 <= nrows);   // block-uniform fast path

    // Stage X tile (zero-fill OOB rows)
    {
        int r  = t >> 4;          // 0..15
        int c0 = (t & 15) * 8;    // 0..120
        const float* xp = X + (long long)(row0 + r) * HID + c0;
        float4 v0, v1;
        if (full) {
            v0 = *(const float4*)(xp);
            v1 = *(const float4*)(xp + 4);
        } else {
            v0 = make_float4(0.f, 0.f, 0.f, 0.f);
            v1 = make_float4(0.f, 0.f, 0.f, 0.f);
            if (row0 + r < nrows) {
                v0 = *(const float4*)(xp);
                v1 = *(const float4*)(xp + 4);
            }
        }
        *(float4*)(&sX[r * SXS + c0])     = v0;
        *(float4*)(&sX[r * SXS + c0 + 4]) = v1;
    }
    // Stage W transposed
    for (int i = t; i < HID * HID; i += 256) {
        int k = i >> 7;
        int n = i & 127;
        sWt[n * SXS + k] = Wm[i];
    }
    __syncthreads();

    const int wave = t >> 5;
    const int lane = t & 31;
    const int m    = lane & 15;           // A row / B col / C col index within tile
    const int koff = (lane >> 4) * 2;     // lanes 16-31 hold K+2,K+3 halves
    const int col  = wave * 16 + m;       // global output column for this lane

    v8f acc = {};
    #pragma unroll 4
    for (int k = 0; k < HID; k += 4) {
        int kb = k + koff;
        v2f a, b;
        a.x = sX[m * SXS + kb];
        a.y = sX[m * SXS + kb + 1];
        b.x = sWt[col * SXS + kb];
        b.y = sWt[col * SXS + kb + 1];
        // D = A(16x4,f32) * B(4x16,f32) + C ; 8-arg form:
        // (neg_a, A, neg_b, B, c_mod, C, reuse_a, reuse_b)
        acc = __builtin_amdgcn_wmma_f32_16x16x4_f32(false, a, false, b,
                                                    (short)0, acc, false, false);
    }

    // C/D layout: VGPR r -> M=r (lanes 0-15) / M=r+8 (lanes 16-31); N = lane%16
    const int mbase = (lane >> 4) * 8;
    float* hp = H + (long long)(row0 + mbase) * HID + col;
    if (full) {
        #pragma unroll
        for (int r = 0; r < 8; ++r) hp[(long long)r * HID] = acc[r];
    } else {
        #pragma unroll
        for (int r = 0; r < 8; ++r) {
            if (row0 + mbase + r < nrows) hp[(long long)r * HID] = acc[r];
        }
    }
}

// ---------------- edge scatter: out[dst] += h[src] * norm ----------------
// 8 edges per block; each 32-lane group moves one 128-float row (float4 per lane).
__global__ __launch_bounds__(256) void scatter_kernel(const float* __restrict__ H,
                                                      const long long* __restrict__ src,
                                                      const long long* __restrict__ dst,
                                                      const float* __restrict__ nrm,
                                                      float* __restrict__ out, long long nE) {
    long long e = (long long)blockIdx.x * 8 + (threadIdx.x >> 5);
    if (e >= nE) return;
    int lane = threadIdx.x & 31;
    int c = lane * 4;
    float w = nrm[e];
    long long s = src[e];
    long long d = dst[e];
    float4 v = *(const float4*)(H + s * HID + c);
    float* o = out + d * HID + c;
    atomicAdd(o + 0, v.x * w);
    atomicAdd(o + 1, v.y * w);
    atomicAdd(o + 2, v.z * w);
    atomicAdd(o + 3, v.w * w);
}

// ---------------- pooling ----------------
__global__ __launch_bounds__(256) void pool_kernel(const float* __restrict__ H,
                                                   float* __restrict__ pool, int nrows) {
    int g = blockIdx.x * 256 + threadIdx.x;
    int c = g & 127;
    int r0 = g >> 7;
    int rstride = (gridDim.x * 256) >> 7;
    float acc = 0.0f;
    for (int r = r0; r < nrows; r += rstride) acc += H[(long long)r * HID + c];
    atomicAdd(&pool[c], acc);
}

__global__ __launch_bounds__(128) void final_kernel(const float* __restrict__ pool,
                                                    const float* __restrict__ lw,
                                                    const float* __restrict__ lb,
                                                    float* __restrict__ out) {
    __shared__ float red[128];
    int t = threadIdx.x;
    red[t] = pool[t] * (1.0f / (float)NNODES) * lw[t];
    __syncthreads();
    #pragma unroll
    for (int s = 64; s > 0; s >>= 1) {
        if (t < s) red[t] += red[t + s];
        __syncthreads();
    }
    if (t == 0) out[0] = red[0] + lb[0];
}

// ---------------- launcher ----------------
extern "C" void kernel_launch(void* const* d_in, const int* in_sizes, int n_in,
                              void* d_out, int out_size, void* d_ws, size_t ws_size,
                              hipStream_t stream) {
    (void)in_sizes; (void)n_in; (void)out_size; (void)ws_size;

    const float*     x    = (const float*)d_in[0];
    const long long* ei   = (const long long*)d_in[1];   // int64 [2, NEDGES]
    const float*     ea   = (const float*)d_in[2];       // [NEDGES, 1]
    const float*     W0   = (const float*)d_in[3];
    const float*     b0   = (const float*)d_in[4];
    const float*     W1   = (const float*)d_in[5];
    const float*     b1   = (const float*)d_in[6];
    const float*     W2   = (const float*)d_in[7];
    const float*     b2   = (const float*)d_in[8];
    const float*     lw   = (const float*)d_in[9];
    const float*     lb   = (const float*)d_in[10];
    float*           outp = (float*)d_out;

    const long long* srcI = ei;
    const long long* dstI = ei + NEDGES;

    // Workspace carve-up (floats)
    float* ws   = (float*)d_ws;
    float* fA   = ws;                                 // node buf A : NNODES*HID
    float* fB   = fA + (long long)NNODES * HID;       // node buf B : NNODES*HID
    float* fNrm = fB + (long long)NNODES * HID;       // per-edge norm : NEDGES
    float* fDeg = fNrm + NEDGES;                      // NNODES
    float* fDnv = fDeg + NNODES;                      // NNODES
    float* fPool= fDnv + NNODES;                      // HID

    const long long nNH = (long long)NNODES * HID;
    const int gNodes  = (NNODES + 255) / 256;
    const int gEdges  = (int)((NEDGES + 255) / 256);
    const int gNH     = (int)((nNH + 255) / 256);
    const int gGemm   = (NNODES + 15) / 16;
    const int gScat   = (int)((NEDGES + 7) / 8);

    // --- normalization (edge-invariant across layers, compute once) ---
    zero_kernel<<<gNodes, 256, 0, stream>>>(fDeg, NNODES);
    deg_kernel<<<gEdges, 256, 0, stream>>>(dstI, ea, fDeg, NEDGES);
    dinv_kernel<<<gNodes, 256, 0, stream>>>(fDeg, fDnv, NNODES);
    norm_kernel<<<gEdges, 256, 0, stream>>>(srcI, dstI, ea, fDnv, fNrm, NEDGES);

    // --- layer 0: x -> fA (gemm) -> fB (scatter, bias-init) ---
    gemm128_wmma<<<gGemm, 256, 0, stream>>>(x, W0, fA, NNODES);
    init_bias_kernel<<<gNH, 256, 0, stream>>>(fB, b0, nNH);
    scatter_kernel<<<gScat, 256, 0, stream>>>(fA, srcI, dstI, fNrm, fB, NEDGES);

    // --- layer 1: fB -> fA (gemm) -> fB (scatter) ---
    gemm128_wmma<<<gGemm, 256, 0, stream>>>(fB, W1, fA, NNODES);
    init_bias_kernel<<<gNH, 256, 0, stream>>>(fB, b1, nNH);
    scatter_kernel<<<gScat, 256, 0, stream>>>(fA, srcI, dstI, fNrm, fB, NEDGES);

    // --- layer 2: fB -> fA (gemm) -> fB (scatter) ---
    gemm128_wmma<<<gGemm, 256, 0, stream>>>(fB, W2, fA, NNODES);
    init_bias_kernel<<<gNH, 256, 0, stream>>>(fB, b2, nNH);
    scatter_kernel<<<gScat, 256, 0, stream>>>(fA, srcI, dstI, fNrm, fB, NEDGES);

    // --- mean pool + linear head ---
    zero_kernel<<<1, 256, 0, stream>>>(fPool, HID);
    pool_kernel<<<2048, 256, 0, stream>>>(fB, fPool, NNODES);
    final_kernel<<<1, 128, 0, stream>>>(fPool, lw, lb, outp);
}